// Sage_76682346102897
// MI455X (gfx1250) — compile-verified
//
#include <hip/hip_runtime.h>
#include <hip/hip_bf16.h>

typedef __attribute__((ext_vector_type(2))) float v2f;
typedef __attribute__((ext_vector_type(8))) float v8f;

#define D_FEAT 64

// ---------------------------------------------------------------------------
// Phase 1: edge scatter. 16 lanes per edge; each lane handles 4 contiguous
// columns with a float4 gather and 4 hardware f32 atomic adds (L2-resident
// destination: neigh_sum is 25.6 MB << 192 MB L2).
// ---------------------------------------------------------------------------
__global__ void sage_scatter_kernel(const float* __restrict__ feat,
                                    const int* __restrict__ src,
                                    const int* __restrict__ dst,
                                    float* __restrict__ nsum,
                                    float* __restrict__ deg,
                                    int n_edges) {
    long long t = (long long)blockIdx.x * blockDim.x + threadIdx.x;
    int e = (int)(t >> 4);        // edge index
    int c = (int)(t & 15);        // column group (4 floats each)
    if (e >= n_edges) return;
    int s = src[e];
    int d = dst[e];
    const float4 v = *(const float4*)(feat + (long long)s * D_FEAT + c * 4);
    float* o = nsum + (long long)d * D_FEAT + c * 4;
    unsafeAtomicAdd(o + 0, v.x);
    unsafeAtomicAdd(o + 1, v.y);
    unsafeAtomicAdd(o + 2, v.z);
    unsafeAtomicAdd(o + 3, v.w);
    if (c == 0) unsafeAtomicAdd(deg + d, 1.0f);
}

// ---------------------------------------------------------------------------
// Phase 2: fused dual-GEMM + bias + ReLU using V_WMMA_F32_16X16X4_F32.
// One wave per 16-node tile. K=64 -> 16 k-steps; N=64 -> 4 n-tiles.
// c = feat @ W_self + (nsum/deg) @ W_neigh, accumulated in one v8f chain.
//
// Per-lane fragment layouts (ISA 7.12.2):
//   A 16x4 f32 : lane m=lane%16, half=lane/16; a[j] = A[m][4*ks + 2*half + j]
//   B  4x16 f32: b[j] = W[4*ks + 2*half + j][ncol],  ncol = nt*16 + m
//   D 16x16 f32: c[r] -> row (r + 8*half), col ncol
// ---------------------------------------------------------------------------
__global__ void sage_wmma_kernel(const float* __restrict__ feat,
                                 const float* __restrict__ nsum,
                                 const float* __restrict__ deg,
                                 const float* __restrict__ Wself,
                                 const float* __restrict__ Wneigh,
                                 const float* __restrict__ bias,
                                 float* __restrict__ out,
                                 int n_nodes) {
    int wave = (int)((blockIdx.x * (long long)blockDim.x + threadIdx.x) >> 5);
    int lane = threadIdx.x & 31;
    int ntiles = (n_nodes + 15) >> 4;
    if (wave >= ntiles) return;          // uniform per wave -> EXEC stays full

    int m    = lane & 15;
    int half = lane >> 4;
    int row  = wave * 16 + m;
    int rclmp = row < n_nodes ? row : (n_nodes - 1);   // clamp tail reads

    float dg  = deg[rclmp];
    float inv = 1.0f / fmaxf(dg, 1.0f);

    const float* frow = feat + (long long)rclmp * D_FEAT;
    const float* nrow = nsum + (long long)rclmp * D_FEAT;

    v2f As[16], An[16];
#pragma unroll
    for (int ks = 0; ks < 16; ++ks) {
        int k = ks * 4 + half * 2;
        As[ks] = *(const v2f*)(frow + k);
        v2f t  = *(const v2f*)(nrow + k);
        An[ks].x = t.x * inv;
        An[ks].y = t.y * inv;
    }

#pragma unroll
    for (int nt = 0; nt < 4; ++nt) {
        int ncol = nt * 16 + m;
        v8f c = {0.f, 0.f, 0.f, 0.f, 0.f, 0.f, 0.f, 0.f};
#pragma unroll
        for (int ks = 0; ks < 16; ++ks) {
            int k = ks * 4 + half * 2;
            v2f bs, bn;
            bs.x = Wself[(k + 0) * D_FEAT + ncol];
            bs.y = Wself[(k + 1) * D_FEAT + ncol];
            bn.x = Wneigh[(k + 0) * D_FEAT + ncol];
            bn.y = Wneigh[(k + 1) * D_FEAT + ncol];
            // 8-arg group: (neg_a, A, neg_b, B, c_mod, C, reuse_a, reuse_b)
            c = __builtin_amdgcn_wmma_f32_16x16x4_f32(false, As[ks], false, bs,
                                                      (short)0, c, false, false);
            c = __builtin_amdgcn_wmma_f32_16x16x4_f32(false, An[ks], false, bn,
                                                      (short)0, c, false, false);
        }
        float b = bias[ncol];
#pragma unroll
        for (int r = 0; r < 8; ++r) {
            int orow = wave * 16 + r + half * 8;
            if (orow < n_nodes) {
                float v = c[r] + b;
                out[(long long)orow * D_FEAT + ncol] = v > 0.f ? v : 0.f;
            }
        }
    }
}

extern "C" void kernel_launch(void* const* d_in, const int* in_sizes, int n_in,
                              void* d_out, int out_size, void* d_ws, size_t ws_size,
                              hipStream_t stream) {
    const float* feat   = (const float*)d_in[0];
    const int*   src    = (const int*)  d_in[1];
    const int*   dst    = (const int*)  d_in[2];
    const float* Wself  = (const float*)d_in[3];
    const float* Wneigh = (const float*)d_in[4];
    const float* bias   = (const float*)d_in[5];
    float*       out    = (float*)d_out;

    int n_nodes = in_sizes[0] / D_FEAT;
    int n_edges = in_sizes[1];

    float* nsum = (float*)d_ws;
    float* deg  = nsum + (size_t)n_nodes * D_FEAT;

    // Zero neigh_sum + deg (async memset is graph-capture safe).
    hipMemsetAsync(d_ws, 0, (size_t)n_nodes * (D_FEAT + 1) * sizeof(float), stream);

    // Edge scatter: 16 lanes per edge.
    long long sthreads = (long long)n_edges * 16;
    int sblocks = (int)((sthreads + 255) / 256);
    sage_scatter_kernel<<<sblocks, 256, 0, stream>>>(feat, src, dst, nsum, deg, n_edges);

    // Fused WMMA GEMM: one wave (32 threads) per 16-node tile.
    int ntiles  = (n_nodes + 15) / 16;
    long long gthreads = (long long)ntiles * 32;
    int gblocks = (int)((gthreads + 255) / 256);
    sage_wmma_kernel<<<gblocks, 256, 0, stream>>>(feat, nsum, deg, Wself, Wneigh,
                                                  bias, out, n_nodes);
}